// GCN_Model_18262200943040
// MI455X (gfx1250) — compile-verified
//
#include <hip/hip_runtime.h>
#include <hip/hip_bf16.h>

typedef float v2f __attribute__((ext_vector_type(2)));
typedef float v8f __attribute__((ext_vector_type(8)));

#define GCN_N       100000
#define GCN_G       500
#define GCN_CIN     128
#define GCN_H       128
#define GCN_COUT    200
#define GCN_NCLS    4
#define BT          256

// ---------------------------------------------------------------------------
// degree / norm precompute
// ---------------------------------------------------------------------------
__global__ void k_deg_init(float* __restrict__ deg, int n) {
    int i = blockIdx.x * blockDim.x + threadIdx.x;
    if (i < n) deg[i] = 1.0f;                       // self-loop weight
}

__global__ void k_deg_scatter(const int* __restrict__ dst,
                              const float* __restrict__ ea,
                              float* __restrict__ deg, int e) {
    int i = blockIdx.x * blockDim.x + threadIdx.x;
    if (i < e) atomicAdd(&deg[dst[i]], ea[i]);
}

__global__ void k_rsqrt(float* __restrict__ dis, int n) {
    int i = blockIdx.x * blockDim.x + threadIdx.x;
    if (i < n) {
        float d = dis[i];
        dis[i] = (d > 0.0f) ? rsqrtf(fmaxf(d, 1e-12f)) : 0.0f;
    }
}

__global__ void k_norm(const int* __restrict__ src, const int* __restrict__ dst,
                       const float* __restrict__ ea, const float* __restrict__ dis,
                       float* __restrict__ norm, int e) {
    int i = blockIdx.x * blockDim.x + threadIdx.x;
    if (i < e) norm[i] = dis[src[i]] * ea[i] * dis[dst[i]];
}

// ---------------------------------------------------------------------------
// weight transpose: Wt[n*K + k] = W[k*Ncol + n]  (<= 200*128 elements, trivial)
// ---------------------------------------------------------------------------
__global__ void k_transpose(const float* __restrict__ W, float* __restrict__ Wt,
                            int K, int Ncol) {
    int idx = blockIdx.x * blockDim.x + threadIdx.x;
    if (idx < K * Ncol) {
        int k = idx / Ncol;
        int n = idx - k * Ncol;
        Wt[(size_t)n * K + k] = W[idx];
    }
}

// ---------------------------------------------------------------------------
// GEMM: T[M x Ncol] = act(A[M x K]) @ W[K x Ncol], V_WMMA_F32_16X16X4_F32.
// Block = 8 waves sharing one 64-column N-range; the block stages its
// 64 x K slice of Wt into LDS once (row stride K+4 words -> lanes hit
// disjoint bank pairs on ds_load_b64). Each wave computes a 16x64 tile:
// per k-step = 1 global float2 (A) + 4 LDS float2 (B) + 4 WMMAs, fully
// unrolled with immediate offsets. Out-of-range B columns are clamped:
// they only feed D columns that are never stored (column n lives in lane n&15).
// ---------------------------------------------------------------------------
template<int K, bool RELU>
__global__ __launch_bounds__(BT)
void k_gemm_wmma(const float* __restrict__ A, const float* __restrict__ Wt,
                 float* __restrict__ T, int M, int Ncol, int tilesN4) {
    const int STR = K + 4;                          // padded LDS row stride (words)
    __shared__ __align__(16) float sW[64 * (K + 4)];

    const int bM   = blockIdx.x / tilesN4;
    const int tN   = blockIdx.x - bM * tilesN4;
    const int n0   = tN * 64;
    const int lane = threadIdx.x & 31;
    const int r    = lane & 15;                     // A row / B,D column within subtile
    const int hi   = lane >> 4;                     // lane-half selects K pair

    // cooperative stage of the 64-column Wt slice (clamped columns)
    for (int t = threadIdx.x; t < 64 * (K / 4); t += BT) {
        int j  = t / (K / 4);
        int k4 = (t - j * (K / 4)) * 4;
        int col = min(n0 + j, Ncol - 1);
        *(float4*)(sW + j * STR + k4) = *(const float4*)(Wt + (size_t)col * K + k4);
    }
    __syncthreads();

    const int tM = bM * 8 + (threadIdx.x >> 5);
    if (tM * 16 >= M) return;                       // wave-uniform: EXEC stays full
    const int m0 = tM * 16;

    v8f c0 = {0.f,0.f,0.f,0.f,0.f,0.f,0.f,0.f};
    v8f c1 = c0, c2 = c0, c3 = c0;

    const float* ap  = A + (size_t)(m0 + r) * K + 2 * hi;
    const float* sb0 = sW + (min(n0 +  0 + r, Ncol - 1) - n0) * STR + 2 * hi;
    const float* sb1 = sW + (min(n0 + 16 + r, Ncol - 1) - n0) * STR + 2 * hi;
    const float* sb2 = sW + (min(n0 + 32 + r, Ncol - 1) - n0) * STR + 2 * hi;
    const float* sb3 = sW + (min(n0 + 48 + r, Ncol - 1) - n0) * STR + 2 * hi;

    #pragma unroll
    for (int k0 = 0; k0 < K; k0 += 4) {
        v2f a = *(const v2f*)(ap + k0);
        if (RELU) { a.x = fmaxf(a.x, 0.0f); a.y = fmaxf(a.y, 0.0f); }
        v2f b0 = *(const v2f*)(sb0 + k0);
        v2f b1 = *(const v2f*)(sb1 + k0);
        v2f b2 = *(const v2f*)(sb2 + k0);
        v2f b3 = *(const v2f*)(sb3 + k0);

        c0 = __builtin_amdgcn_wmma_f32_16x16x4_f32(false, a, false, b0, (short)0, c0, false, false);
        c1 = __builtin_amdgcn_wmma_f32_16x16x4_f32(false, a, false, b1, (short)0, c1, false, false);
        c2 = __builtin_amdgcn_wmma_f32_16x16x4_f32(false, a, false, b2, (short)0, c2, false, false);
        c3 = __builtin_amdgcn_wmma_f32_16x16x4_f32(false, a, false, b3, (short)0, c3, false, false);
    }

    // store: D row = m0 + hi*8 + i, col = n0 + j*16 + r (guard real columns)
    const int mbase = m0 + hi * 8;
    float* tb = T + (size_t)mbase * Ncol;
    v8f acc[4] = {c0, c1, c2, c3};
    #pragma unroll
    for (int j = 0; j < 4; ++j) {
        int nc = n0 + j * 16 + r;
        if (nc < Ncol) {
            #pragma unroll
            for (int i = 0; i < 8; ++i)
                tb[(size_t)i * Ncol + nc] = acc[j][i];
        }
    }
}

// ---------------------------------------------------------------------------
// H[i,c] = b[c] + T[i,c] * dis[i]^2   (self-loop term + bias; full init of H)
// ---------------------------------------------------------------------------
__global__ void k_selfloop_bias(const float* __restrict__ T, const float* __restrict__ dis,
                                const float* __restrict__ bias, float* __restrict__ H,
                                int n, int C) {
    long long idx = (long long)blockIdx.x * blockDim.x + threadIdx.x;
    long long tot = (long long)n * C;
    if (idx < tot) {
        int i = (int)(idx / C);
        int c = (int)(idx - (long long)i * C);
        float d = dis[i];
        H[idx] = bias[c] + T[idx] * d * d;
    }
}

// ---------------------------------------------------------------------------
// edge scatter: H[dst] += T[src] * norm — one wave32 per edge.
// Edge scalars forced uniform (readfirstlane) -> scalar loads; float4 gather.
// ---------------------------------------------------------------------------
__global__ __launch_bounds__(BT)
void k_scatter(const float* __restrict__ T, const int* __restrict__ src,
               const int* __restrict__ dst, const float* __restrict__ norm,
               float* __restrict__ H, int e, int C) {
    int wave = __builtin_amdgcn_readfirstlane(blockIdx.x * (BT / 32) + (threadIdx.x >> 5));
    int lane = threadIdx.x & 31;
    if (wave >= e) return;
    int   s = src[wave];
    int   d = dst[wave];
    float w = norm[wave];
    const float4* tin = (const float4*)(T + (size_t)s * C);   // rows are 16B aligned
    float*       hout = H + (size_t)d * C;
    for (int c4 = lane; c4 * 4 < C; c4 += 32) {
        float4 v = tin[c4];
        int c = c4 * 4;
        atomicAdd(&hout[c + 0], v.x * w);
        atomicAdd(&hout[c + 1], v.y * w);
        atomicAdd(&hout[c + 2], v.z * w);
        atomicAdd(&hout[c + 3], v.w * w);
    }
}

// ---------------------------------------------------------------------------
// pooling
// ---------------------------------------------------------------------------
__global__ void k_pool_zero(float* __restrict__ ps, float* __restrict__ cnt,
                            int g, int C) {
    int i = blockIdx.x * blockDim.x + threadIdx.x;
    if (i < g * C) ps[i] = 0.0f;
    if (i < g) cnt[i] = 0.0f;
}

__global__ void k_pool_accum(const float* __restrict__ H, const int* __restrict__ batch,
                             float* __restrict__ ps, float* __restrict__ cnt,
                             int n, int C) {
    long long idx = (long long)blockIdx.x * blockDim.x + threadIdx.x;
    long long tot = (long long)n * C;
    if (idx < tot) {
        int i = (int)(idx / C);
        int c = (int)(idx - (long long)i * C);
        int g = batch[i];
        atomicAdd(&ps[(size_t)g * C + c], H[idx]);
        if (c == 0) atomicAdd(&cnt[g], 1.0f);
    }
}

__global__ void k_final(const float* __restrict__ ps, const float* __restrict__ cnt,
                        const float* __restrict__ Wp, const float* __restrict__ bp,
                        float* __restrict__ out, int g, int C, int K) {
    int idx = blockIdx.x * blockDim.x + threadIdx.x;
    if (idx >= g * K) return;
    int gi = idx / K, k = idx - gi * K;
    float inv = 1.0f / fmaxf(cnt[gi], 1.0f);
    const float* row = ps + (size_t)gi * C;
    float acc = bp[k];
    for (int c = 0; c < C; ++c)
        acc += (row[c] * inv) * Wp[(size_t)c * K + k];
    out[idx] = acc;
}

// ---------------------------------------------------------------------------
// launcher
// ---------------------------------------------------------------------------
static inline int cdiv(long long a, long long b) { return (int)((a + b - 1) / b); }

extern "C" void kernel_launch(void* const* d_in, const int* in_sizes, int n_in,
                              void* d_out, int out_size, void* d_ws, size_t ws_size,
                              hipStream_t stream) {
    const float* x    = (const float*)d_in[0];
    const int*   ei   = (const int*)d_in[1];
    const float* ea   = (const float*)d_in[2];
    const int*   batch= (const int*)d_in[3];
    const float* W0   = (const float*)d_in[4];
    const float* b0   = (const float*)d_in[5];
    const float* W1   = (const float*)d_in[6];
    const float* b1   = (const float*)d_in[7];
    const float* W2   = (const float*)d_in[8];
    const float* b2   = (const float*)d_in[9];
    const float* Wp   = (const float*)d_in[10];
    const float* bp   = (const float*)d_in[11];
    float* out = (float*)d_out;

    const int N = in_sizes[0] / GCN_CIN;
    const int E = in_sizes[1] / 2;
    const int G = GCN_G;
    const int* src = ei;            // edge_index[0]
    const int* dst = ei + E;        // edge_index[1]

    // workspace layout (floats)
    float* ws   = (float*)d_ws;
    float* dis  = ws;                               // N
    float* norm = dis + N;                          // E
    float* bufT = norm + E;                         // N * 200
    float* bufH = bufT + (size_t)N * GCN_COUT;      // N * 200
    float* ps   = bufH + (size_t)N * GCN_COUT;      // G * 200
    float* cnt  = ps + (size_t)G * GCN_COUT;        // G
    float* wt   = cnt + G;                          // 200 * 128 (transposed weights)

    // --- normalization coefficients ---
    k_deg_init<<<cdiv(N, BT), BT, 0, stream>>>(dis, N);
    k_deg_scatter<<<cdiv(E, BT), BT, 0, stream>>>(dst, ea, dis, E);
    k_rsqrt<<<cdiv(N, BT), BT, 0, stream>>>(dis, N);
    k_norm<<<cdiv(E, BT), BT, 0, stream>>>(src, dst, ea, dis, norm, E);

    const int tilesM  = cdiv(N, 16);
    const int blocksM = cdiv(tilesM, 8);            // 8 M-tiles (waves) per block

    // --- layer 0: T = x @ W0 ; H = selfloop+bias ; scatter edges ---
    {
        int tilesN4 = cdiv(GCN_H, 64);
        k_transpose<<<cdiv(GCN_CIN * GCN_H, BT), BT, 0, stream>>>(W0, wt, GCN_CIN, GCN_H);
        k_gemm_wmma<GCN_CIN, false><<<blocksM * tilesN4, BT, 0, stream>>>(x, wt, bufT, N, GCN_H, tilesN4);
        k_selfloop_bias<<<cdiv((long long)N * GCN_H, BT), BT, 0, stream>>>(bufT, dis, b0, bufH, N, GCN_H);
        k_scatter<<<cdiv(E, BT / 32), BT, 0, stream>>>(bufT, src, dst, norm, bufH, E, GCN_H);
    }
    // --- layer 1: T = relu(H) @ W1 ---
    {
        int tilesN4 = cdiv(GCN_H, 64);
        k_transpose<<<cdiv(GCN_H * GCN_H, BT), BT, 0, stream>>>(W1, wt, GCN_H, GCN_H);
        k_gemm_wmma<GCN_H, true><<<blocksM * tilesN4, BT, 0, stream>>>(bufH, wt, bufT, N, GCN_H, tilesN4);
        k_selfloop_bias<<<cdiv((long long)N * GCN_H, BT), BT, 0, stream>>>(bufT, dis, b1, bufH, N, GCN_H);
        k_scatter<<<cdiv(E, BT / 32), BT, 0, stream>>>(bufT, src, dst, norm, bufH, E, GCN_H);
    }
    // --- layer 2: T = relu(H) @ W2 (no activation after) ---
    {
        int tilesN4 = cdiv(GCN_COUT, 64);
        k_transpose<<<cdiv(GCN_H * GCN_COUT, BT), BT, 0, stream>>>(W2, wt, GCN_H, GCN_COUT);
        k_gemm_wmma<GCN_H, true><<<blocksM * tilesN4, BT, 0, stream>>>(bufH, wt, bufT, N, GCN_COUT, tilesN4);
        k_selfloop_bias<<<cdiv((long long)N * GCN_COUT, BT), BT, 0, stream>>>(bufT, dis, b2, bufH, N, GCN_COUT);
        k_scatter<<<cdiv(E, BT / 32), BT, 0, stream>>>(bufT, src, dst, norm, bufH, E, GCN_COUT);
    }

    // --- mean pool + classifier ---
    k_pool_zero<<<cdiv(G * GCN_COUT, BT), BT, 0, stream>>>(ps, cnt, G, GCN_COUT);
    k_pool_accum<<<cdiv((long long)N * GCN_COUT, BT), BT, 0, stream>>>(bufH, batch, ps, cnt, N, GCN_COUT);
    k_final<<<cdiv(G * GCN_NCLS, BT), BT, 0, stream>>>(ps, cnt, Wp, bp, out, G, GCN_COUT, GCN_NCLS);
}